// GCN_214748365116
// MI455X (gfx1250) — compile-verified
//
#include <hip/hip_runtime.h>

#define N_NODESX 50000
#define N_EDGESX 800000
#define HIDX 64
#define OUTDX 10
#define NGRAPHX 128
#define NLAYERX 5

typedef float v2f __attribute__((ext_vector_type(2)));
typedef float v8f __attribute__((ext_vector_type(8)));

// ---------------- degree / norm ----------------
__global__ void deg_kernel(const int* __restrict__ dst, int* __restrict__ deg, int E) {
    int e = blockIdx.x * blockDim.x + threadIdx.x;
    if (e < E) atomicAdd(&deg[dst[e]], 1);
}

__global__ void dinv_kernel(const int* __restrict__ deg, float* __restrict__ dinv, int N) {
    int n = blockIdx.x * blockDim.x + threadIdx.x;
    if (n < N) dinv[n] = rsqrtf((float)(deg[n] + 1));  // +1 self loop; deg>=1 always
}

// ---------------- fp32 WMMA GEMM: C[nrows x 64] = A[nrows x 64] * W[64 x 64] ----------------
// one wave -> 16 rows x 64 cols (four 16x16 tiles), K unrolled in steps of 4
__global__ __launch_bounds__(128) void gemm64_wmma(const float* __restrict__ A,
                                                   const float* __restrict__ W,
                                                   float* __restrict__ C, int nrows) {
    const int lane = threadIdx.x & 31;
    const int wave = threadIdx.x >> 5;
    const int tile = blockIdx.x * 4 + wave;           // 16-row tile index
    if (tile * 16 >= nrows) return;                   // wave-uniform: EXEC stays all-ones
    const int col  = lane & 15;
    const int koff = (lane >> 4) << 1;                // 0 for lanes 0-15, 2 for lanes 16-31
    const float* arow = A + (size_t)(tile * 16 + col) * HIDX;

    v8f acc0 = {}, acc1 = {}, acc2 = {}, acc3 = {};
#pragma unroll
    for (int k = 0; k < HIDX; k += 4) {
        v2f a;                                        // A frag: row=col, K=k+koff, k+koff+1
        a.x = arow[k + koff];
        a.y = arow[k + koff + 1];
        const float* wb = W + (size_t)(k + koff) * HIDX + col;  // B frag: K rows of W
        v2f b0; b0.x = wb[0];       b0.y = wb[HIDX];
        v2f b1; b1.x = wb[16];      b1.y = wb[HIDX + 16];
        v2f b2; b2.x = wb[32];      b2.y = wb[HIDX + 32];
        v2f b3; b3.x = wb[48];      b3.y = wb[HIDX + 48];
        acc0 = __builtin_amdgcn_wmma_f32_16x16x4_f32(false, a, false, b0, (short)0, acc0, false, false);
        acc1 = __builtin_amdgcn_wmma_f32_16x16x4_f32(false, a, false, b1, (short)0, acc1, false, false);
        acc2 = __builtin_amdgcn_wmma_f32_16x16x4_f32(false, a, false, b2, (short)0, acc2, false, false);
        acc3 = __builtin_amdgcn_wmma_f32_16x16x4_f32(false, a, false, b3, (short)0, acc3, false, false);
    }
    // D layout: VGPR j -> M = tile*16 + 8*(lane>=16) + j, N = n0 + (lane&15)
    const int mbase = tile * 16 + ((lane >> 4) << 3);
#pragma unroll
    for (int j = 0; j < 8; ++j) {
        float* crow = C + (size_t)(mbase + j) * HIDX + col;
        crow[0]  = acc0[j];
        crow[16] = acc1[j];
        crow[32] = acc2[j];
        crow[48] = acc3[j];
    }
}

// ---------------- self-loop init: h_next = hw * dinv^2 + b ----------------
__global__ void init_kernel(const float* __restrict__ hw, const float* __restrict__ dinv,
                            const float* __restrict__ b, float* __restrict__ hnx) {
    int i = blockIdx.x * blockDim.x + threadIdx.x;
    if (i >= N_NODESX * HIDX) return;
    int n = i >> 6, c = i & 63;
    float di = dinv[n];
    hnx[i] = hw[i] * di * di + b[c];
}

// ---------------- edge scatter: 16 threads per edge, float4 per thread ----------------
__global__ void scatter_kernel(const float* __restrict__ hw, const float* __restrict__ dinv,
                               const int* __restrict__ src, const int* __restrict__ dst,
                               float* __restrict__ hnx, int E) {
    long long t = (long long)blockIdx.x * blockDim.x + threadIdx.x;
    int e = (int)(t >> 4);
    if (e >= E) return;
    int cg = ((int)t & 15) << 2;
    int s = src[e], d = dst[e];
    float nrm = dinv[s] * dinv[d];
    const float4 v = *(const float4*)(hw + (size_t)s * HIDX + cg);
    float* o = hnx + (size_t)d * HIDX + cg;
    atomicAdd(o + 0, v.x * nrm);
    atomicAdd(o + 1, v.y * nrm);
    atomicAdd(o + 2, v.z * nrm);
    atomicAdd(o + 3, v.w * nrm);
}

// ---------------- batch-norm statistics: stats[0..63]=sum, [64..127]=sumsq ----------------
__global__ void stats_kernel(const float* __restrict__ h, float* __restrict__ stats) {
    int c = threadIdx.x & 63;
    int row = blockIdx.x * 4 + (threadIdx.x >> 6);
    float s = 0.f, q = 0.f;
    for (; row < N_NODESX; row += gridDim.x * 4) {
        float v = h[(size_t)row * HIDX + c];
        s += v; q += v * v;
    }
    atomicAdd(&stats[c], s);
    atomicAdd(&stats[HIDX + c], q);
}

__global__ void bn_relu_kernel(const float* __restrict__ hnx, const float* __restrict__ stats,
                               const float* __restrict__ gamma, const float* __restrict__ beta,
                               float* __restrict__ h) {
    int i = blockIdx.x * blockDim.x + threadIdx.x;
    if (i >= N_NODESX * HIDX) return;
    int c = i & 63;
    const float invN = 1.0f / (float)N_NODESX;
    float mu  = stats[c] * invN;
    float var = stats[HIDX + c] * invN - mu * mu;   // biased var, matches jnp.var
    float inv = rsqrtf(var + 1e-5f);
    float v = (hnx[i] - mu) * inv * gamma[c] + beta[c];
    h[i] = v > 0.f ? v : 0.f;
}

// ---------------- global_add_pool ----------------
__global__ void pool_kernel(const float* __restrict__ h, const int* __restrict__ batch,
                            float* __restrict__ pooled) {
    int i = blockIdx.x * blockDim.x + threadIdx.x;
    if (i >= N_NODESX * HIDX) return;
    int n = i >> 6, c = i & 63;
    atomicAdd(&pooled[batch[n] * HIDX + c], h[i]);
}

// ---------------- score += pooled @ W_pred + b_pred ----------------
__global__ void pred_kernel(const float* __restrict__ pooled, const float* __restrict__ Wp,
                            const float* __restrict__ bp, float* __restrict__ score) {
    int t = blockIdx.x * blockDim.x + threadIdx.x;
    if (t >= NGRAPHX * OUTDX) return;
    int g = t / OUTDX, o = t % OUTDX;
    float acc = bp[o];
    const float* pg = pooled + (size_t)g * HIDX;
#pragma unroll
    for (int c = 0; c < HIDX; ++c) acc += pg[c] * Wp[c * OUTDX + o];
    score[t] += acc;
}

extern "C" void kernel_launch(void* const* d_in, const int* in_sizes, int n_in,
                              void* d_out, int out_size, void* d_ws, size_t ws_size,
                              hipStream_t stream) {
    (void)in_sizes; (void)n_in; (void)out_size; (void)ws_size;
    const float* x      = (const float*)d_in[0];
    const int*   eidx   = (const int*)d_in[1];
    const int*   batch  = (const int*)d_in[2];
    const float* W_gcn  = (const float*)d_in[3];
    const float* b_gcn  = (const float*)d_in[4];
    const float* gamma  = (const float*)d_in[5];
    const float* beta   = (const float*)d_in[6];
    const float* W_pred = (const float*)d_in[7];
    const float* b_pred = (const float*)d_in[8];
    float* score = (float*)d_out;

    const int* src = eidx;               // edge_index row 0
    const int* dst = eidx + N_EDGESX;    // edge_index row 1

    char* ws = (char*)d_ws;
    size_t off = 0;
    auto alloc = [&](size_t bytes) -> void* {
        void* p = ws + off;
        off += (bytes + 255) & ~(size_t)255;
        return p;
    };
    int*   deg    = (int*)  alloc((size_t)N_NODESX * 4);
    float* dinv   = (float*)alloc((size_t)N_NODESX * 4);
    float* hw     = (float*)alloc((size_t)N_NODESX * HIDX * 4);
    float* h      = (float*)alloc((size_t)N_NODESX * HIDX * 4);
    float* hnx    = (float*)alloc((size_t)N_NODESX * HIDX * 4);
    float* stats  = (float*)alloc(2 * HIDX * 4);
    float* pooled = (float*)alloc((size_t)NGRAPHX * HIDX * 4);

    const int NC = N_NODESX * HIDX;                 // 3.2M elements
    const int blkNC = (NC + 255) / 256;

    hipMemsetAsync(deg, 0, (size_t)N_NODESX * 4, stream);
    hipMemsetAsync(score, 0, (size_t)NGRAPHX * OUTDX * 4, stream);

    deg_kernel<<<(N_EDGESX + 255) / 256, 256, 0, stream>>>(dst, deg, N_EDGESX);
    dinv_kernel<<<(N_NODESX + 255) / 256, 256, 0, stream>>>(deg, dinv, N_NODESX);

    // hidden_rep[0] = x : pool + predict with layer-0 head
    hipMemsetAsync(pooled, 0, (size_t)NGRAPHX * HIDX * 4, stream);
    pool_kernel<<<blkNC, 256, 0, stream>>>(x, batch, pooled);
    pred_kernel<<<(NGRAPHX * OUTDX + 255) / 256, 256, 0, stream>>>(
        pooled, W_pred, b_pred, score);

    const float* hin = x;
    for (int l = 0; l < NLAYERX - 1; ++l) {
        // h @ W_gcn[l]  (WMMA fp32)
        gemm64_wmma<<<(N_NODESX + 63) / 64, 128, 0, stream>>>(
            hin, W_gcn + (size_t)l * HIDX * HIDX, hw, N_NODESX);
        // self-loop contribution + bias (also fully initializes the accumulator)
        init_kernel<<<blkNC, 256, 0, stream>>>(hw, dinv, b_gcn + (size_t)l * HIDX, hnx);
        // edge message scatter
        scatter_kernel<<<((long long)N_EDGESX * 16 + 255) / 256, 256, 0, stream>>>(
            hw, dinv, src, dst, hnx, N_EDGESX);
        // batch-norm (training stats) + relu
        hipMemsetAsync(stats, 0, 2 * HIDX * 4, stream);
        stats_kernel<<<512, 256, 0, stream>>>(hnx, stats);
        bn_relu_kernel<<<blkNC, 256, 0, stream>>>(
            hnx, stats, gamma + (size_t)l * HIDX, beta + (size_t)l * HIDX, h);
        // pool + predict with head l+1
        hipMemsetAsync(pooled, 0, (size_t)NGRAPHX * HIDX * 4, stream);
        pool_kernel<<<blkNC, 256, 0, stream>>>(h, batch, pooled);
        pred_kernel<<<(NGRAPHX * OUTDX + 255) / 256, 256, 0, stream>>>(
            pooled, W_pred + (size_t)(l + 1) * HIDX * OUTDX, b_pred + (size_t)(l + 1) * OUTDX, score);
        hin = h;
    }
}